// GIN_42949672960516
// MI455X (gfx1250) — compile-verified
//
#include <hip/hip_runtime.h>

#define N_NODES  100000
#define N_EDGES  1000000
#define N_GRAPHS 256
#define D        64
#define BN_EPS   1e-5f

typedef __attribute__((ext_vector_type(2))) float v2f;
typedef __attribute__((ext_vector_type(8))) float v8f;

// ---------------------------------------------------------------- zero fill
__global__ void zero_kernel(float4* __restrict__ p, int n4) {
    int i = blockIdx.x * blockDim.x + threadIdx.x;
    int stride = gridDim.x * blockDim.x;
    for (; i < n4; i += stride) p[i] = make_float4(0.f, 0.f, 0.f, 0.f);
}

// ------------------------------------------------- edge scatter: agg += x[src]
// one wave per edge; lane covers feature f and f+32 (coalesced 128B+128B)
__global__ void edge_scatter_kernel(const float* __restrict__ x,
                                    const int* __restrict__ ei,
                                    float* __restrict__ agg) {
    int wid  = (blockIdx.x * blockDim.x + threadIdx.x) >> 5;
    int lane = threadIdx.x & 31;
    if (wid >= N_EDGES) return;                    // wave-uniform
    int src = ei[wid];
    int dst = ei[N_EDGES + wid];
    const float* xs = x + (size_t)src * D;
    float*       ad = agg + (size_t)dst * D;
    atomicAdd(ad + lane,       xs[lane]);          // L2-resident f32 atomics
    atomicAdd(ad + lane + 32,  xs[lane + 32]);
}

// ---------------------------------------------------------------- GEMM #1
// T = (X [+ AGG]) @ W + bias   (rows x 64 @ 64 x 64), fused column sum/sumsq
// for BatchNorm. One wave computes a 16-row strip via V_WMMA_F32_16X16X4_F32.
__global__ void __launch_bounds__(256)
gemm_bias_stats_kernel(const float* __restrict__ X, const float* __restrict__ AGG,
                       const float* __restrict__ W, const float* __restrict__ bias,
                       float* __restrict__ T, float* __restrict__ stats, int rows) {
    __shared__ float ldsW[64 * 64];   // 16 KB weight tile
    __shared__ float ldsSum[64];
    __shared__ float ldsSq[64];
    int tid = threadIdx.x;
    for (int i = tid; i < 64 * 64; i += 256) ldsW[i] = W[i];
    if (tid < 64) { ldsSum[tid] = 0.f; ldsSq[tid] = 0.f; }
    __syncthreads();

    int wid  = tid >> 5;
    int lane = tid & 31;
    int m0   = (blockIdx.x * 8 + wid) * 16;
    if (m0 < rows) {                               // wave-uniform: EXEC stays full
        int mrow = m0 + (lane & 15);
        int koff = (lane >> 4) * 2;                // ISA 16x4 f32 A layout
        int n    = lane & 15;

        v2f a[16];
        const float* xr = X + (size_t)mrow * D;
        const float* ar = AGG ? AGG + (size_t)mrow * D : nullptr;
#pragma unroll
        for (int s = 0; s < 16; ++s) {
            int c = 4 * s + koff;
            float2 xv = *(const float2*)(xr + c);
            if (ar) { float2 av = *(const float2*)(ar + c); xv.x += av.x; xv.y += av.y; }
            a[s].x = xv.x; a[s].y = xv.y;
        }
#pragma unroll
        for (int nt = 0; nt < 4; ++nt) {
            float bv = bias[nt * 16 + n];          // bias depends on column only
            v8f c;
#pragma unroll
            for (int r = 0; r < 8; ++r) c[r] = bv;
#pragma unroll
            for (int s = 0; s < 16; ++s) {         // K = 64 in 16 steps of 4
                v2f b;
                b.x = ldsW[(4 * s + koff)     * 64 + nt * 16 + n];
                b.y = ldsW[(4 * s + koff + 1) * 64 + nt * 16 + n];
                c = __builtin_amdgcn_wmma_f32_16x16x4_f32(
                        false, a[s], false, b, (short)0, c, false, false);
            }
            float ssum = 0.f, ssq = 0.f;
            int rbase = m0 + ((lane >> 4) << 3);   // lanes 16-31 hold rows M=8..15
#pragma unroll
            for (int r = 0; r < 8; ++r) {
                float v = c[r];
                T[(size_t)(rbase + r) * D + nt * 16 + n] = v;
                ssum += v; ssq += v * v;
            }
            // lanes l and l^16 share the same column -> combine halves
            ssum += __shfl_xor(ssum, 16, 32);
            ssq  += __shfl_xor(ssq,  16, 32);
            if (lane < 16) {
                atomicAdd(&ldsSum[nt * 16 + n], ssum);
                atomicAdd(&ldsSq [nt * 16 + n], ssq);
            }
        }
    }
    __syncthreads();
    if (tid < 64) {
        atomicAdd(&stats[tid],      ldsSum[tid]);
        atomicAdd(&stats[64 + tid], ldsSq[tid]);
    }
}

// --------------------------------------------------------- BN fold (64 thr)
__global__ void bn_finalize_kernel(const float* __restrict__ stats,
                                   const float* __restrict__ gamma,
                                   const float* __restrict__ beta,
                                   float* __restrict__ scale,
                                   float* __restrict__ shift, float inv_count) {
    int c = threadIdx.x;
    float mu  = stats[c] * inv_count;
    float var = stats[64 + c] * inv_count - mu * mu;   // biased var (torch BN)
    float sc  = gamma[c] * rsqrtf(var + BN_EPS);
    scale[c] = sc;
    shift[c] = beta[c] - mu * sc;
}

// ---------------------------------------------------------------- GEMM #2
// Xout = relu( relu(T*scale+shift) @ W + bias )
__global__ void __launch_bounds__(256)
gemm_affine_relu_kernel(const float* __restrict__ T, const float* __restrict__ scale,
                        const float* __restrict__ shift, const float* __restrict__ W,
                        const float* __restrict__ bias, float* __restrict__ Xout,
                        int rows) {
    __shared__ float ldsW[64 * 64];
    __shared__ float ldsScale[64];
    __shared__ float ldsShift[64];
    int tid = threadIdx.x;
    for (int i = tid; i < 64 * 64; i += 256) ldsW[i] = W[i];
    if (tid < 64) { ldsScale[tid] = scale[tid]; ldsShift[tid] = shift[tid]; }
    __syncthreads();

    int wid  = tid >> 5;
    int lane = tid & 31;
    int m0   = (blockIdx.x * 8 + wid) * 16;
    if (m0 >= rows) return;                        // no barrier after this: safe

    int mrow = m0 + (lane & 15);
    int koff = (lane >> 4) * 2;
    int n    = lane & 15;

    v2f a[16];
    const float* tr = T + (size_t)mrow * D;
#pragma unroll
    for (int s = 0; s < 16; ++s) {
        int c = 4 * s + koff;
        float2 tv = *(const float2*)(tr + c);
        a[s].x = fmaxf(tv.x * ldsScale[c]     + ldsShift[c],     0.f);
        a[s].y = fmaxf(tv.y * ldsScale[c + 1] + ldsShift[c + 1], 0.f);
    }
#pragma unroll
    for (int nt = 0; nt < 4; ++nt) {
        float bv = bias[nt * 16 + n];
        v8f c;
#pragma unroll
        for (int r = 0; r < 8; ++r) c[r] = bv;
#pragma unroll
        for (int s = 0; s < 16; ++s) {
            v2f b;
            b.x = ldsW[(4 * s + koff)     * 64 + nt * 16 + n];
            b.y = ldsW[(4 * s + koff + 1) * 64 + nt * 16 + n];
            c = __builtin_amdgcn_wmma_f32_16x16x4_f32(
                    false, a[s], false, b, (short)0, c, false, false);
        }
        int rbase = m0 + ((lane >> 4) << 3);
#pragma unroll
        for (int r = 0; r < 8; ++r)
            Xout[(size_t)(rbase + r) * D + nt * 16 + n] = fmaxf(c[r], 0.f);
    }
}

// -------------------------------------------------------- global add pool
__global__ void pool_kernel(const float* __restrict__ x, const int* __restrict__ batch,
                            float* __restrict__ pooled) {
    int idx = blockIdx.x * blockDim.x + threadIdx.x;
    if (idx >= N_NODES * D) return;
    int node = idx >> 6;
    int f    = idx & 63;
    atomicAdd(&pooled[(size_t)batch[node] * D + f], x[idx]);
}

// ------------------------------------------- head: [256,64] -> [256,2]
__global__ void final_out_kernel(const float* __restrict__ T, const float* __restrict__ scale,
                                 const float* __restrict__ shift, const float* __restrict__ W2,
                                 const float* __restrict__ b2, float* __restrict__ out) {
    int row = threadIdx.x;                         // 256 threads, 1 block
    float acc0 = b2[0], acc1 = b2[1];
    const float* tr = T + (size_t)row * D;
#pragma unroll
    for (int c = 0; c < 64; ++c) {
        float h = fmaxf(tr[c] * scale[c] + shift[c], 0.f);
        acc0 += h * W2[c * 2 + 0];
        acc1 += h * W2[c * 2 + 1];
    }
    out[row * 2 + 0] = acc0;
    out[row * 2 + 1] = acc1;
}

extern "C" void kernel_launch(void* const* d_in, const int* in_sizes, int n_in,
                              void* d_out, int out_size, void* d_ws, size_t ws_size,
                              hipStream_t stream) {
    (void)in_sizes; (void)n_in; (void)out_size; (void)ws_size;
    const float* x     = (const float*)d_in[0];
    const int*   ei    = (const int*)  d_in[1];
    const int*   batch = (const int*)  d_in[2];
    const float* cW1   = (const float*)d_in[3];
    const float* cb1   = (const float*)d_in[4];
    const float* cg    = (const float*)d_in[5];
    const float* cbt   = (const float*)d_in[6];
    const float* cW2   = (const float*)d_in[7];
    const float* cb2   = (const float*)d_in[8];
    const float* mW1   = (const float*)d_in[9];
    const float* mb1   = (const float*)d_in[10];
    const float* mg    = (const float*)d_in[11];
    const float* mbt   = (const float*)d_in[12];
    const float* mW2   = (const float*)d_in[13];
    const float* mb2   = (const float*)d_in[14];
    float* out = (float*)d_out;

    const size_t NF = (size_t)N_NODES * D;         // 6.4M floats
    float* ws    = (float*)d_ws;
    float* A     = ws;                             // node buf / agg (ping)
    float* B     = ws + NF;                        // node buf / agg (pong)
    float* T     = ws + 2 * NF;                    // pre-BN activations
    float* stats = ws + 3 * NF;                    // sum[64], sumsq[64]
    float* scale = stats + 128;
    float* shift = stats + 192;
    float* pooled = out + N_GRAPHS * 2;            // second tuple output

    const int ZB = 256;
    const int nodeBlocks  = 782;                   // ceil(6250 strips / 8 waves)
    const int edgeBlocks  = (N_EDGES * 32 + 255) / 256;
    const int poolBlocks  = (N_NODES * D + 255) / 256;
    const float invN = 1.0f / (float)N_NODES;
    const float invG = 1.0f / (float)N_GRAPHS;

    // zero the pooled/out region (harness poisons d_out)
    zero_kernel<<<(N_GRAPHS * (D + 2) / 4 + ZB - 1) / ZB, ZB, 0, stream>>>(
        (float4*)out, N_GRAPHS * (D + 2) / 4);

    const float* xin = x;
    float* aggbuf[3] = {A, B, A};
    for (int l = 0; l < 3; ++l) {
        float* agg = aggbuf[l];
        zero_kernel<<<1024, ZB, 0, stream>>>((float4*)agg, (int)(NF / 4));
        zero_kernel<<<1, 32, 0, stream>>>((float4*)stats, 32);
        edge_scatter_kernel<<<edgeBlocks, ZB, 0, stream>>>(xin, ei, agg);
        gemm_bias_stats_kernel<<<nodeBlocks, ZB, 0, stream>>>(
            xin, agg, cW1 + l * 4096, cb1 + l * 64, T, stats, N_NODES);
        bn_finalize_kernel<<<1, 64, 0, stream>>>(
            stats, cg + l * 64, cbt + l * 64, scale, shift, invN);
        gemm_affine_relu_kernel<<<nodeBlocks, ZB, 0, stream>>>(
            T, scale, shift, cW2 + l * 4096, cb2 + l * 64, agg, N_NODES);
        xin = agg;                                 // Xout aliases agg (dead after GEMM1)
    }

    // global_add_pool into the output buffer itself
    pool_kernel<<<poolBlocks, ZB, 0, stream>>>(xin, batch, pooled);

    // head MLP: reuse the WMMA GEMM+stats kernel for pooled @ mW1 + mb1
    zero_kernel<<<1, 32, 0, stream>>>((float4*)stats, 32);
    gemm_bias_stats_kernel<<<2, ZB, 0, stream>>>(
        pooled, nullptr, mW1, mb1, T, stats, N_GRAPHS);
    bn_finalize_kernel<<<1, 64, 0, stream>>>(stats, mg, mbt, scale, shift, invG);
    final_out_kernel<<<1, 256, 0, stream>>>(T, scale, shift, mW2, mb2, out);
}